// NeuralGeodesicFlows_26603027431828
// MI455X (gfx1250) — compile-verified
//
#include <hip/hip_runtime.h>
#include <math.h>

// Neural geodesic flow integrator for MI455X (gfx1250, wave32).
// One wave = 32 independent geodesics. The two 64x64 matvecs per RHS eval are
// batched across the wave's 32 points and executed as V_WMMA_F32_16X16X4_F32
// GEMMs (fp32 end-to-end, matching the reference precision). Small 8-dim
// per-point linear algebra (g = A A^T + eps I, Cholesky solve) is per-lane.
// tanh is a branchless v_exp_f32/v_rcp_f32 sequence (~6e-8 abs error) to keep
// the VALU critical path short and EXEC all-ones.

typedef float v2f __attribute__((ext_vector_type(2)));
typedef float v8f __attribute__((ext_vector_type(8)));

static __device__ __forceinline__ v8f wmma4(v2f a, v2f b, v8f c) {
  // D(16x16,f32) = A(16x4,f32) * B(4x16,f32) + C
  return __builtin_amdgcn_wmma_f32_16x16x4_f32(
      false, a, false, b, (short)0, c, false, false);
}

// Branchless tanh: sign(x) * (1 - e)/(1 + e), e = exp(-2|x|).
// One v_exp_f32 + one v_rcp_f32; abs error ~1 ulp of 0.5 for all x.
static __device__ __forceinline__ float fast_tanh(float x) {
  const float ax = fabsf(x);
  const float e = __expf(-2.0f * ax);
  const float r = __builtin_amdgcn_rcpf(1.0f + e);
  const float t = (1.0f - e) * r;
  return copysignf(t, x);
}

#define LT(i, j) ((i) * ((i) + 1) / 2 + (j))

__global__ __launch_bounds__(32)
void NeuralGeodesicFlows_kernel(const float* __restrict__ zin,
                                const float* __restrict__ tin,
                                const float* __restrict__ W1,
                                const float* __restrict__ b1,
                                const float* __restrict__ W2,
                                const float* __restrict__ b2,
                                float* __restrict__ out) {
  // ---- LDS staging ----
  __shared__ __align__(16) float W2s[64 * 64];   // row-major W2[n][m]
  __shared__ __align__(16) float W1Ts[64 * 16];  // W1^T zero-padded to 16 cols
  __shared__ __align__(16) float Xs[32 * 8];     // per-point x
  __shared__ __align__(16) float Hs[32 * 64];    // per-point h = tanh(xW1+b1)
  __shared__ __align__(16) float As[32 * 64];    // per-point A flat, reused as VU
  __shared__ __align__(16) float DCs[32 * 64];   // per-point d .* c
  __shared__ __align__(16) float Ss[32 * 16];    // per-point S = W1^T (d.*c)

  const int tid = threadIdx.x;   // 0..31
  const int lo = tid & 15;       // column lane within 16
  const int hi = tid >> 4;       // half-select (K/row split in frags)
  const int gp = blockIdx.x * 32 + tid;

  // Prefetch weights (global_prefetch_b8 path) ahead of the staging loads.
  for (int i = tid; i < 128; i += 32)
    __builtin_prefetch((const char*)W2 + (size_t)i * 128, 0, 1);
  if (tid < 16) __builtin_prefetch((const char*)W1 + (size_t)tid * 128, 0, 1);

  // Stage weights into LDS (once).
  for (int i = tid; i < 64 * 64 / 4; i += 32)
    ((float4*)W2s)[i] = ((const float4*)W2)[i];
  for (int r = tid; r < 64; r += 32) {
#pragma unroll
    for (int c = 0; c < 16; ++c)
      W1Ts[r * 16 + c] = (c < 8) ? W1[c * 64 + r] : 0.0f;
  }

  // Per-lane register-resident B-fragments of W1 (for GEMM0) and biases.
  v2f w1b[2][4];
  float b1v[4], b2v[4];
#pragma unroll
  for (int k = 0; k < 2; ++k) {
#pragma unroll
    for (int nt = 0; nt < 4; ++nt) {
      const int r0 = 4 * k + 2 * hi;
      const int c = nt * 16 + lo;
      w1b[k][nt].x = W1[r0 * 64 + c];
      w1b[k][nt].y = W1[(r0 + 1) * 64 + c];
    }
  }
#pragma unroll
  for (int nt = 0; nt < 4; ++nt) {
    b1v[nt] = b1[nt * 16 + lo];
    b2v[nt] = b2[nt * 16 + lo];
  }
  __syncthreads();

  // ---- load state ----
  float zx[8], zv[8];
  {
    const float* zr = zin + (size_t)gp * 16;
#pragma unroll
    for (int i = 0; i < 8; ++i) { zx[i] = zr[i]; zv[i] = zr[8 + i]; }
  }
  const float tval = tin[gp];
  float nsf = ceilf(fabsf(tval) * 10.0f);
  nsf = fminf(fmaxf(nsf, 5.0f), 150.0f);
  const int ns = (int)nsf;
  const float dt = tval / nsf;

  // Wave-uniform step bound (keeps EXEC all-ones around WMMA).
  int wmax = ns;
#pragma unroll
  for (int off = 16; off > 0; off >>= 1) {
    int o = __shfl_xor(wmax, off, 32);
    wmax = (o > wmax) ? o : wmax;
  }

  // Cholesky factor persists between factorization and solve within one eval.
  float Lm[36], dinv[8];

  // ---- RHS eval: dv = -solve(g + eps I, W1^T((1-h^2) .* (W2 vec(v u^T)))) ----
  auto f_eval = [&](const float* x, const float* vv, float* dv) {
    // scatter x
#pragma unroll
    for (int q = 0; q < 8; ++q) Xs[tid * 8 + q] = x[q];
    __syncthreads();

    // GEMM0: pre = X(32x8) @ W1(8x64) + b1 ; h = tanh(pre)
#pragma unroll
    for (int hf = 0; hf < 2; ++hf) {
#pragma unroll
      for (int nt = 0; nt < 4; ++nt) {
        v8f acc;
#pragma unroll
        for (int r = 0; r < 8; ++r) acc[r] = b1v[nt];
#pragma unroll
        for (int k = 0; k < 2; ++k) {
          const float* xp = &Xs[(hf * 16 + lo) * 8 + 4 * k + 2 * hi];
          v2f a; a.x = xp[0]; a.y = xp[1];
          acc = wmma4(a, w1b[k][nt], acc);
        }
        const int col = nt * 16 + lo;
#pragma unroll
        for (int r = 0; r < 8; ++r) {
          const int row = hf * 16 + hi * 8 + r;
          Hs[row * 64 + col] = fast_tanh(acc[r]);
        }
      }
    }
    __syncthreads();

    // GEMM1: Aflat(32x64) = H(32x64) @ W2(64x64) + b2
#pragma unroll
    for (int hf = 0; hf < 2; ++hf) {
#pragma unroll
      for (int nt = 0; nt < 4; ++nt) {
        v8f acc;
#pragma unroll
        for (int r = 0; r < 8; ++r) acc[r] = b2v[nt];
        for (int k = 0; k < 16; ++k) {
          const int kb = 4 * k + 2 * hi;
          const float* hp = &Hs[(hf * 16 + lo) * 64 + kb];
          v2f a; a.x = hp[0]; a.y = hp[1];
          v2f b;
          b.x = W2s[kb * 64 + nt * 16 + lo];
          b.y = W2s[(kb + 1) * 64 + nt * 16 + lo];
          acc = wmma4(a, b, acc);
        }
        const int col = nt * 16 + lo;
#pragma unroll
        for (int r = 0; r < 8; ++r)
          As[(hf * 16 + hi * 8 + r) * 64 + col] = acc[r];
      }
    }
    __syncthreads();

    // Per-lane: g = A A^T + eps I, Cholesky, u = A^T v, scatter vu^T.
    {
      float Ar[64];
#pragma unroll
      for (int i = 0; i < 64; ++i) Ar[i] = As[tid * 64 + i];

#pragma unroll
      for (int i = 0; i < 8; ++i) {
#pragma unroll
        for (int j = 0; j <= i; ++j) {
          float sum = (i == j) ? 1e-5f : 0.0f;
#pragma unroll
          for (int q = 0; q < 8; ++q) sum += Ar[i * 8 + q] * Ar[j * 8 + q];
#pragma unroll
          for (int k2 = 0; k2 < j; ++k2)
            sum -= Lm[LT(i, k2)] * Lm[LT(j, k2)];
          if (j == i) {
            const float l = sqrtf(sum);
            Lm[LT(i, i)] = l;
            dinv[i] = 1.0f / l;
          } else {
            Lm[LT(i, j)] = sum * dinv[j];
          }
        }
      }

      float u[8];
#pragma unroll
      for (int q = 0; q < 8; ++q) {
        float s = 0.0f;
#pragma unroll
        for (int i = 0; i < 8; ++i) s += vv[i] * Ar[i * 8 + q];
        u[q] = s;
      }
#pragma unroll
      for (int a = 0; a < 8; ++a)
#pragma unroll
        for (int q = 0; q < 8; ++q)
          As[tid * 64 + a * 8 + q] = vv[a] * u[q];  // reuse As as VU
    }
    __syncthreads();

    // GEMM2: C(32x64) = VU(32x64) @ W2^T ; fuse dc = (1-h^2) .* c
#pragma unroll
    for (int hf = 0; hf < 2; ++hf) {
#pragma unroll
      for (int nt = 0; nt < 4; ++nt) {
        v8f acc;
#pragma unroll
        for (int r = 0; r < 8; ++r) acc[r] = 0.0f;
        for (int k = 0; k < 16; ++k) {
          const int kb = 4 * k + 2 * hi;
          const float* vp = &As[(hf * 16 + lo) * 64 + kb];
          v2f a; a.x = vp[0]; a.y = vp[1];
          const float* wp = &W2s[(nt * 16 + lo) * 64 + kb];  // W2^T frag: contiguous
          v2f b; b.x = wp[0]; b.y = wp[1];
          acc = wmma4(a, b, acc);
        }
        const int col = nt * 16 + lo;
#pragma unroll
        for (int r = 0; r < 8; ++r) {
          const int row = hf * 16 + hi * 8 + r;
          const float h = Hs[row * 64 + col];
          DCs[row * 64 + col] = (1.0f - h * h) * acc[r];
        }
      }
    }
    __syncthreads();

    // GEMM3: S(32x8) = DC(32x64) @ W1^T(64x8, zero-padded to 16 cols)
#pragma unroll
    for (int hf = 0; hf < 2; ++hf) {
      v8f acc;
#pragma unroll
      for (int r = 0; r < 8; ++r) acc[r] = 0.0f;
      for (int k = 0; k < 16; ++k) {
        const int kb = 4 * k + 2 * hi;
        const float* dp = &DCs[(hf * 16 + lo) * 64 + kb];
        v2f a; a.x = dp[0]; a.y = dp[1];
        v2f b;
        b.x = W1Ts[kb * 16 + lo];
        b.y = W1Ts[(kb + 1) * 16 + lo];
        acc = wmma4(a, b, acc);
      }
#pragma unroll
      for (int r = 0; r < 8; ++r)
        Ss[(hf * 16 + hi * 8 + r) * 16 + lo] = acc[r];
    }
    __syncthreads();

    // Per-lane: dv = -solve(L L^T, S)
    {
      float sv[8], w[8], y[8];
#pragma unroll
      for (int i = 0; i < 8; ++i) sv[i] = Ss[tid * 16 + i];
#pragma unroll
      for (int i = 0; i < 8; ++i) {
        float acc = sv[i];
#pragma unroll
        for (int j = 0; j < i; ++j) acc -= Lm[LT(i, j)] * w[j];
        w[i] = acc * dinv[i];
      }
#pragma unroll
      for (int i = 7; i >= 0; --i) {
        float acc = w[i];
#pragma unroll
        for (int j = i + 1; j < 8; ++j) acc -= Lm[LT(j, i)] * y[j];
        y[i] = acc * dinv[i];
      }
#pragma unroll
      for (int k = 0; k < 8; ++k) dv[k] = -y[k];
    }
    __syncthreads();
  };

  // ---- RK4 time loop (wave-uniform trip count, per-lane masked commit) ----
  for (int step = 0; step < wmax; ++step) {
    float yx[8], yv[8], accx[8], accv[8], k1x[8], k1v[8];
#pragma unroll
    for (int i = 0; i < 8; ++i) {
      yx[i] = zx[i]; yv[i] = zv[i];
      accx[i] = 0.0f; accv[i] = 0.0f;
    }
    bool bad = false;

#pragma unroll 1
    for (int s = 0; s < 4; ++s) {
      float dv[8], kx[8], kv[8];
      f_eval(yx, yv, dv);
#pragma unroll
      for (int i = 0; i < 8; ++i) { kx[i] = yv[i]; kv[i] = dv[i]; }
#pragma unroll
      for (int i = 0; i < 8; ++i) {
        bad = bad || !(fabsf(kx[i]) <= 3.0e38f) || !(fabsf(kv[i]) <= 3.0e38f);
      }
      if (s == 0) {
#pragma unroll
        for (int i = 0; i < 8; ++i) { k1x[i] = kx[i]; k1v[i] = kv[i]; }
      }
      const float bw = (s == 0 || s == 3) ? 1.0f : 2.0f;
#pragma unroll
      for (int i = 0; i < 8; ++i) { accx[i] += bw * kx[i]; accv[i] += bw * kv[i]; }
      if (s < 3) {
        const float aw = ((s == 2) ? 1.0f : 0.5f) * dt;
#pragma unroll
        for (int i = 0; i < 8; ++i) {
          yx[i] = zx[i] + aw * kx[i];
          yv[i] = zv[i] + aw * kv[i];
        }
      }
    }

    const float c6 = dt * (1.0f / 6.0f);
    const bool active = step < ns;
#pragma unroll
    for (int i = 0; i < 8; ++i) {
      const float fx = zx[i] + c6 * accx[i];
      const float fv = zv[i] + c6 * accv[i];
      const float ex = zx[i] + dt * k1x[i];
      const float ev = zv[i] + dt * k1v[i];
      const float nx = bad ? ex : fx;
      const float nv = bad ? ev : fv;
      zx[i] = active ? nx : zx[i];
      zv[i] = active ? nv : zv[i];
    }
  }

  // ---- store ----
  {
    float* orow = out + (size_t)gp * 16;
#pragma unroll
    for (int i = 0; i < 8; ++i) { orow[i] = zx[i]; orow[8 + i] = zv[i]; }
  }
}

extern "C" void kernel_launch(void* const* d_in, const int* in_sizes, int n_in,
                              void* d_out, int out_size, void* d_ws, size_t ws_size,
                              hipStream_t stream) {
  const float* z  = (const float*)d_in[0];
  const float* t  = (const float*)d_in[1];
  const float* W1 = (const float*)d_in[2];
  const float* b1 = (const float*)d_in[3];
  const float* W2 = (const float*)d_in[4];
  const float* b2 = (const float*)d_in[5];
  float* out = (float*)d_out;

  const int npts = in_sizes[0] / 16;  // BS = 8192
  dim3 grid(npts / 32), block(32);
  hipLaunchKernelGGL(NeuralGeodesicFlows_kernel, grid, block, 0, stream,
                     z, t, W1, b1, W2, b2, out);
}